// Tracker_86045374808494
// MI455X (gfx1250) — compile-verified
//
#include <hip/hip_runtime.h>
#include <math.h>

// ---------------- problem constants ----------------
#define S_LEN   8
#define NTRK    256
#define DCH     128          // d latent channels per plane
#define CTOT    384          // 3*DCH channels in feature planes
#define CD      7
#define CD2     49           // 7*7
#define CD4     2401         // 49*49
#define K_PAD   2432         // CD4 padded up to multiple of 32 (and 64)
#define H1      384
#define H2      256
#define GELU_C  0.7978845608028654f

typedef __attribute__((ext_vector_type(2))) float v2f;
typedef __attribute__((ext_vector_type(8))) float v8f;

// ---------------- avg pool 2x2 (pyramid) ----------------
__global__ __launch_bounds__(256) void avgpool2(const float* __restrict__ in,
                                                float* __restrict__ out,
                                                int v_in, int total) {
  int idx = blockIdx.x * 256 + threadIdx.x;
  if (idx >= total) return;
  int vo = v_in >> 1;
  int x  = idx % vo;
  int y  = (idx / vo) % vo;
  int fc = idx / (vo * vo);
  const float* p = in + (size_t)fc * v_in * v_in;
  int r0 = (2 * y) * v_in + 2 * x;
  int r1 = r0 + v_in;
  out[idx] = 0.25f * (p[r0] + p[r0 + 1] + p[r1] + p[r1 + 1]);
}

// ---------------- zero c4 pad columns ----------------
__global__ __launch_bounds__(256) void zero_c4_pad(float* __restrict__ c4) {
  int idx = blockIdx.x * 256 + threadIdx.x;
  const int padc = K_PAD - CD4;                 // 31
  if (idx >= S_LEN * NTRK * padc) return;
  int row = idx / padc;
  int c   = idx % padc;
  c4[(size_t)row * K_PAD + CD4 + c] = 0.f;
}

// ---------------- device bilinear (border clamp, align_corners) ----------------
__device__ __forceinline__ float bilin(const float* __restrict__ img, int v,
                                       float sx, float sy) {
  float mx = (float)(v - 1);
  sx = fminf(fmaxf(sx, 0.f), mx);
  sy = fminf(fmaxf(sy, 0.f), mx);
  float x0f = floorf(sx), y0f = floorf(sy);
  int x0 = (int)x0f, y0 = (int)y0f;
  int x1 = min(x0 + 1, v - 1), y1 = min(y0 + 1, v - 1);
  float wx = sx - x0f, wy = sy - y0f;
  float p00 = img[y0 * v + x0], p01 = img[y0 * v + x1];
  float p10 = img[y1 * v + x0], p11 = img[y1 * v + x1];
  return p00 * (1.f - wy) * (1.f - wx) + p01 * (1.f - wy) * wx +
         p10 * wy * (1.f - wx)        + p11 * wy * wx;
}

// ---------------- TF sampling, K-pair interleaved -------------------
// Layout: tf[n][k/2][xy(64pad)][k&1]  ->  B fragments are single b64 LDS loads.
// trilinear with t = 0 (B = 1)  ->  bilinear on frame 0 of the level.
__global__ __launch_bounds__(128) void tf_sample(const float* __restrict__ lvl,
                                                 const float* __restrict__ ci,
                                                 float* __restrict__ tf,
                                                 int v, int a, int b, int plane,
                                                 float inv_scale) {
  int bid = blockIdx.x;
  int n = bid >> 6, o = bid & 63;
  int di = threadIdx.x;                               // 0..127 (K index)
  float* dst = tf + (((size_t)n * 64 + (di >> 1)) * 64 + o) * 2 + (di & 1);
  if (o >= CD2) { *dst = 0.f; return; }               // zero pad cols 49..63
  int ih = o / CD, iw = o % CD;                       // h offsets x, w offsets y
  float sx = ci[n * 3 + a] * inv_scale + (float)(ih - 3);
  float sy = ci[n * 3 + b] * inv_scale + (float)(iw - 3);
  const float* img = lvl + (size_t)(di * 3 + plane) * v * v;  // frame 0
  *dst = bilin(img, v, sx, sy);
}

// ---------------- per-(s,n) correlation: C(49x49) = CF(49x128) @ TF(128x49) ----
__global__ __launch_bounds__(128) void corr_einsum(const float* __restrict__ lvl,
                                                   const float* __restrict__ coords,
                                                   const float* __restrict__ tf,
                                                   float* __restrict__ c4,
                                                   int v, int a, int b, int plane,
                                                   float inv_scale) {
  __shared__ __align__(16) float cf[64][132];    // A: rows 49..63 zero-filled
  __shared__ __align__(16) float tl[64][128];    // B pairs: [k/2][xy*2 + (k&1)]
  const int tid = threadIdx.x;
  const int sn  = blockIdx.x;                    // s*256 + n
  const int s   = sn >> 8;
  const int n   = sn & 255;

  // stage TF (8192 floats, already pair-interleaved in global)
  {
    const float4* src4 = (const float4*)(tf + (size_t)n * DCH * 64);
    float4* dst4 = (float4*)&tl[0][0];
    for (int i = tid; i < DCH * 64 / 4; i += 128) dst4[i] = src4[i];
  }
  // sample CF: 64*128 values (rows >= 49 written as zero)
  const float fx = coords[(size_t)sn * 3 + a] * inv_scale;
  const float fy = coords[(size_t)sn * 3 + b] * inv_scale;
  for (int idx = tid; idx < 64 * DCH; idx += 128) {
    int hw = idx >> 7;                           // /128
    int di = idx & 127;
    float val = 0.f;
    if (hw < CD2) {
      int ih = hw / CD, iw = hw % CD;
      const float* img = lvl + (size_t)(s * CTOT + di * 3 + plane) * v * v;
      val = bilin(img, v, fx + (float)(ih - 3), fy + (float)(iw - 3));
    }
    cf[hw][di] = val;
  }
  __syncthreads();

  const int w    = tid >> 5;                     // wave 0..3 -> 16-row strip
  const int lane = tid & 31;
  const int lh   = lane >> 4;                    // lane half selects K pair
  const int ll   = lane & 15;
  const int row  = w * 16 + ll;                  // A row (hw), zero rows >= 49

  v8f acc[4] = {v8f{}, v8f{}, v8f{}, v8f{}};
#pragma unroll 2
  for (int k = 0; k < DCH; k += 4) {
    int k0 = k + 2 * lh;                         // even -> 8B aligned
    v2f af = *(const v2f*)&cf[row][k0];
    int kp = (k >> 1) + lh;
#pragma unroll
    for (int nt = 0; nt < 4; ++nt) {
      v2f bf = *(const v2f*)&tl[kp][2 * (nt * 16 + ll)];
      acc[nt] = __builtin_amdgcn_wmma_f32_16x16x4_f32(
          false, af, false, bf, (short)0, acc[nt], false, false);
    }
  }

  // epilogue: unconditional LDS round-trip, then linear coalesced copy-out.
  __syncthreads();                               // done reading cf everywhere
  float* ct = &cf[0][0];                         // reuse as 64x64 C tile, stride 68
#pragma unroll
  for (int nt = 0; nt < 4; ++nt) {
    int xy = nt * 16 + ll;
#pragma unroll
    for (int r = 0; r < 8; ++r) {
      int hw2 = w * 16 + 8 * lh + r;
      ct[hw2 * 68 + xy] = acc[nt][r];
    }
  }
  __syncthreads();
  float* crow = c4 + (size_t)sn * K_PAD;         // col = hw*49 + xy, contiguous
  for (int idx = tid; idx < CD4; idx += 128) {
    int hw = idx / CD2, xy = idx - hw * CD2;
    crow[idx] = ct[hw * 68 + xy];
  }
}

// ---------------- MLP layer 1: H = gelu(C4 @ W1 + b1) ----------------
// M=2048, N=384, K=2432 (A pad cols are zero; B guarded past 2401)
__global__ __launch_bounds__(256) void mlp1_gemm(const float* __restrict__ c4,
                                                 const float* __restrict__ w1,
                                                 const float* __restrict__ b1,
                                                 float* __restrict__ h) {
  __shared__ __align__(16) float As[64][36];
  __shared__ __align__(16) float Bs[32][68];
  const int tid = threadIdx.x;
  const int bm = blockIdx.x, bn = blockIdx.y;
  const int w = tid >> 5, lane = tid & 31, lh = lane >> 4, ll = lane & 15;
  const int mt = w & 3, ntb = (w >> 2) << 1;     // 16 tiles / 8 waves
  const int rowA0 = bm * 64, colB0 = bn * 64;
  v8f acc[2] = {v8f{}, v8f{}};

  for (int kc = 0; kc < K_PAD; kc += 32) {
    __syncthreads();
#pragma unroll
    for (int it = 0; it < 2; ++it) {             // A: 64x32
      int idx = tid + it * 256;
      int r = idx >> 3, q = idx & 7;
      float4 val = *(const float4*)(c4 + (size_t)(rowA0 + r) * K_PAD + kc + q * 4);
      *(float4*)&As[r][q * 4] = val;
    }
#pragma unroll
    for (int it = 0; it < 2; ++it) {             // B: 32x64 (guarded)
      int idx = tid + it * 256;
      int kr = idx >> 4, q = idx & 15;
      int kg = kc + kr;
      float4 val = make_float4(0.f, 0.f, 0.f, 0.f);
      if (kg < CD4) val = *(const float4*)(w1 + (size_t)kg * H1 + colB0 + q * 4);
      *(float4*)&Bs[kr][q * 4] = val;
    }
    __syncthreads();
#pragma unroll
    for (int kk = 0; kk < 32; kk += 4) {
      int k0 = kk + 2 * lh;                      // even -> 8B aligned
      v2f af = *(const v2f*)&As[mt * 16 + ll][k0];
#pragma unroll
      for (int j = 0; j < 2; ++j) {
        int nc = (ntb + j) * 16 + ll;
        v2f bf;
        bf.x = Bs[k0][nc];
        bf.y = Bs[k0 + 1][nc];
        acc[j] = __builtin_amdgcn_wmma_f32_16x16x4_f32(
            false, af, false, bf, (short)0, acc[j], false, false);
      }
    }
  }
#pragma unroll
  for (int j = 0; j < 2; ++j) {
    int col = colB0 + (ntb + j) * 16 + ll;
    float bias = b1[col];
#pragma unroll
    for (int r = 0; r < 8; ++r) {
      int row = rowA0 + mt * 16 + 8 * lh + r;
      float x = acc[j][r] + bias;
      float t = tanhf(GELU_C * (x + 0.044715f * x * x * x));
      h[(size_t)row * H1 + col] = 0.5f * x * (1.f + t);
    }
  }
}

// ---------------- MLP layer 2: out[..,lev*256+c] (+)= H @ W2 + b2 ------------
__global__ __launch_bounds__(256) void mlp2_gemm(const float* __restrict__ h,
                                                 const float* __restrict__ w2,
                                                 const float* __restrict__ b2,
                                                 float* __restrict__ out,
                                                 int level, int accum) {
  __shared__ __align__(16) float As[64][36];
  __shared__ __align__(16) float Bs[32][68];
  const int tid = threadIdx.x;
  const int bm = blockIdx.x, bn = blockIdx.y;
  const int w = tid >> 5, lane = tid & 31, lh = lane >> 4, ll = lane & 15;
  const int mt = w & 3, ntb = (w >> 2) << 1;
  const int rowA0 = bm * 64, colB0 = bn * 64;
  v8f acc[2] = {v8f{}, v8f{}};

  for (int kc = 0; kc < H1; kc += 32) {
    __syncthreads();
#pragma unroll
    for (int it = 0; it < 2; ++it) {             // A: 64x32 from h (stride 384)
      int idx = tid + it * 256;
      int r = idx >> 3, q = idx & 7;
      float4 val = *(const float4*)(h + (size_t)(rowA0 + r) * H1 + kc + q * 4);
      *(float4*)&As[r][q * 4] = val;
    }
#pragma unroll
    for (int it = 0; it < 2; ++it) {             // B: 32x64 from w2 (stride 256)
      int idx = tid + it * 256;
      int kr = idx >> 4, q = idx & 15;
      float4 val = *(const float4*)(w2 + (size_t)(kc + kr) * H2 + colB0 + q * 4);
      *(float4*)&Bs[kr][q * 4] = val;
    }
    __syncthreads();
#pragma unroll
    for (int kk = 0; kk < 32; kk += 4) {
      int k0 = kk + 2 * lh;
      v2f af = *(const v2f*)&As[mt * 16 + ll][k0];
#pragma unroll
      for (int j = 0; j < 2; ++j) {
        int nc = (ntb + j) * 16 + ll;
        v2f bf;
        bf.x = Bs[k0][nc];
        bf.y = Bs[k0 + 1][nc];
        acc[j] = __builtin_amdgcn_wmma_f32_16x16x4_f32(
            false, af, false, bf, (short)0, acc[j], false, false);
      }
    }
  }
#pragma unroll
  for (int j = 0; j < 2; ++j) {
    int col = colB0 + (ntb + j) * 16 + ll;
    float bias = b2[col];
#pragma unroll
    for (int r = 0; r < 8; ++r) {
      int row = rowA0 + mt * 16 + 8 * lh + r;
      float val = acc[j][r] + bias;
      size_t oi = (size_t)row * 1024 + (size_t)level * 256 + col;
      if (accum) out[oi] += val;
      else out[oi] = val;
    }
  }
}

// ---------------- host-side orchestration ----------------
extern "C" void kernel_launch(void* const* d_in, const int* in_sizes, int n_in,
                              void* d_out, int out_size, void* d_ws, size_t ws_size,
                              hipStream_t stream) {
  (void)in_sizes; (void)n_in; (void)out_size; (void)ws_size;
  const float* coords      = (const float*)d_in[0];  // (1,8,256,3)
  const float* coords_init = (const float*)d_in[1];  // (1,256,3)
  const float* fplanes     = (const float*)d_in[2];  // (8,384,96,96)
  const float* w1          = (const float*)d_in[3];  // (2401,384)
  const float* b1          = (const float*)d_in[4];  // (384)
  const float* w2          = (const float*)d_in[5];  // (384,256)
  const float* b2          = (const float*)d_in[6];  // (256)
  float* out = (float*)d_out;                        // (1,8,256,1024)

  float* ws   = (float*)d_ws;
  float* pyr1 = ws;                                  // 8*384*48*48
  float* pyr2 = pyr1 + (size_t)8 * 384 * 48 * 48;    // 8*384*24*24
  float* pyr3 = pyr2 + (size_t)8 * 384 * 24 * 24;    // 8*384*12*12
  float* tf   = pyr3 + (size_t)8 * 384 * 12 * 12;    // 256*128*64 (pair-interleaved)
  float* c4   = tf   + (size_t)NTRK * DCH * 64;      // 2048*2432
  float* hbuf = c4   + (size_t)S_LEN * NTRK * K_PAD; // 2048*384

  // feature pyramid
  { int tot = 8 * 384 * 48 * 48;
    avgpool2<<<(tot + 255) / 256, 256, 0, stream>>>(fplanes, pyr1, 96, tot); }
  { int tot = 8 * 384 * 24 * 24;
    avgpool2<<<(tot + 255) / 256, 256, 0, stream>>>(pyr1, pyr2, 48, tot); }
  { int tot = 8 * 384 * 12 * 12;
    avgpool2<<<(tot + 255) / 256, 256, 0, stream>>>(pyr2, pyr3, 24, tot); }
  // zero the padded tail of each c4 row (read-only afterwards)
  { int tot = S_LEN * NTRK * (K_PAD - CD4);
    zero_c4_pad<<<(tot + 255) / 256, 256, 0, stream>>>(c4); }

  const float* lvl_ptr[4] = {fplanes, pyr1, pyr2, pyr3};
  const int    lvl_v[4]   = {96, 48, 24, 12};
  const int pa[3] = {0, 0, 1};
  const int pb[3] = {1, 2, 2};

  for (int lev = 0; lev < 4; ++lev) {
    float inv = 1.0f / (float)(1 << lev);
    for (int k = 0; k < 3; ++k) {
      tf_sample<<<NTRK * 64, 128, 0, stream>>>(
          lvl_ptr[lev], coords_init, tf, lvl_v[lev], pa[k], pb[k], k, inv);
      corr_einsum<<<S_LEN * NTRK, 128, 0, stream>>>(
          lvl_ptr[lev], coords, tf, c4, lvl_v[lev], pa[k], pb[k], k, inv);
      dim3 g1(32, 6);
      mlp1_gemm<<<g1, 256, 0, stream>>>(c4, w1, b1, hbuf);
      dim3 g2(32, 4);
      mlp2_gemm<<<g2, 256, 0, stream>>>(hbuf, w2, b2, out, lev, k > 0 ? 1 : 0);
    }
  }
}